// Rod_Block_24601572671767
// MI455X (gfx1250) — compile-verified
//
#include <hip/hip_runtime.h>
#include <math.h>

#define C_INC  64
#define C_OUTC 128
#define G_GRP  4
#define CG_    16
#define P_TAP  9
#define N_IMG  8
#define H_IMG  64
#define W_IMG  64
#define HW_IMG 4096
#define PIX_T  32768
#define LN_EPS 1e-6f
#define BN_EPS 1e-5f

typedef __attribute__((ext_vector_type(2))) float v2f;
typedef __attribute__((ext_vector_type(8))) float v8f;

// ---------------------------------------------------------------------------
// 16x16 f32 GEMM tile via V_WMMA_F32_16X16X4_F32, K = 64 (16 WMMA ops)
// A: row-major [M x K] (lda), B: row-major [K x N] (ldb)
// ---------------------------------------------------------------------------
__device__ __forceinline__ v8f wmma_gemm_tile_k64(const float* __restrict__ A, int lda,
                                                  const float* __restrict__ B, int ldb,
                                                  v8f acc) {
  const int lane = threadIdx.x & 31;
  const int half = lane >> 4;     // 0: K+{0,1}, 1: K+{2,3}
  const int l16  = lane & 15;
  const float* ap = A + l16 * lda + 2 * half;      // A row = l16
  const float* bp = B + (2 * half) * ldb + l16;    // B col = l16
#pragma unroll
  for (int k0 = 0; k0 < C_INC; k0 += 4) {
    v2f a, b;
    a.x = ap[k0 + 0];
    a.y = ap[k0 + 1];
    b.x = bp[(k0 + 0) * ldb];
    b.y = bp[(k0 + 1) * ldb];
    acc = __builtin_amdgcn_wmma_f32_16x16x4_f32(false, a, false, b, (short)0, acc,
                                                false, false);
  }
  return acc;
}

// ---------------------------------------------------------------------------
// 1) Global average pool over HW per (n, c)
// ---------------------------------------------------------------------------
__global__ void k_pool(const float* __restrict__ x, float* __restrict__ pooled) {
  __shared__ float red[256];
  const int nc  = blockIdx.x;          // 0..511
  const int tid = threadIdx.x;
  const float* src = x + (size_t)nc * HW_IMG;
  float s = 0.f;
  for (int i = tid; i < HW_IMG; i += 256) s += src[i];
  red[tid] = s;
  __syncthreads();
  for (int off = 128; off > 0; off >>= 1) {
    if (tid < off) red[tid] += red[tid + off];
    __syncthreads();
  }
  if (tid == 0) pooled[nc] = red[0] * (1.0f / HW_IMG);
}

// ---------------------------------------------------------------------------
// 2) SE MLP: gain = sigmoid(relu(pooled@w1+b1)@w2+b2)   (8x64 -> tiny)
// ---------------------------------------------------------------------------
__global__ void k_gain_mlp(const float* __restrict__ pooled,
                           const float* __restrict__ w1, const float* __restrict__ b1,
                           const float* __restrict__ w2, const float* __restrict__ b2,
                           float* __restrict__ gain) {
  __shared__ float sh[16];
  const int tid = threadIdx.x;  // 64 threads
  for (int n = 0; n < N_IMG; ++n) {
    if (tid < 16) {
      float a = b1[tid];
      for (int c = 0; c < C_INC; ++c) a += pooled[n * C_INC + c] * w1[c * 16 + tid];
      sh[tid] = fmaxf(a, 0.f);
    }
    __syncthreads();
    float a = b2[tid];
    for (int j = 0; j < 16; ++j) a += sh[j] * w2[j * C_INC + tid];
    gain[n * C_INC + tid] = 1.0f / (1.0f + expf(-a));
    __syncthreads();
  }
}

// ---------------------------------------------------------------------------
// 3) xh[p][c] = x*(1+gain) + relu(refl*tap_w + tap_b)   (NCHW -> NHWC)
// ---------------------------------------------------------------------------
__global__ void k_build_xh(const float* __restrict__ x, const float* __restrict__ refl,
                           const float* __restrict__ gain,
                           const float* __restrict__ tap_w, const float* __restrict__ tap_b,
                           float* __restrict__ xh) {
  const int idx = blockIdx.x * blockDim.x + threadIdx.x;  // 2,097,152
  const int c = idx & 63;
  const int p = idx >> 6;
  const int w = p & 63;
  const int h = (p >> 6) & 63;
  const int n = p >> 12;
  const float xv = x[(((size_t)n * C_INC + c) * HW_IMG) + (h << 6) + w];
  const float rv = refl[(size_t)n * HW_IMG + (h << 6) + w];
  const float rc = fmaxf(rv * tap_w[c] + tap_b[c], 0.f);
  xh[idx] = xv * (1.0f + gain[n * C_INC + c]) + rc;
}

// ---------------------------------------------------------------------------
// 4) input_proj: xp = xh @ inp_w + inp_b   (WMMA, M=32768,N=64,K=64)
// ---------------------------------------------------------------------------
__global__ void k_input_proj(const float* __restrict__ xh, const float* __restrict__ w,
                             const float* __restrict__ bias, float* __restrict__ xp) {
  const int wid  = blockIdx.x * (blockDim.x >> 5) + (threadIdx.x >> 5);
  const int tm   = wid >> 2;      // 2048 tile rows
  const int tn   = wid & 3;       // 4 tile cols
  const int lane = threadIdx.x & 31;
  const int half = lane >> 4, l16 = lane & 15;
  const float bv = bias[tn * 16 + l16];
  v8f acc;
#pragma unroll
  for (int r = 0; r < 8; ++r) acc[r] = bv;
  acc = wmma_gemm_tile_k64(xh + (size_t)tm * 16 * C_INC, C_INC, w + tn * 16, C_INC, acc);
  float* out = xp + (size_t)tm * 16 * C_INC + tn * 16 + l16;
#pragma unroll
  for (int r = 0; r < 8; ++r) out[(r + half * 8) * C_INC] = acc[r];
}

// ---------------------------------------------------------------------------
// 5) depthwise 3x3 (SAME) + bias + LayerNorm(channels) + exact GELU
// ---------------------------------------------------------------------------
__global__ void k_dw_ln_gelu(const float* __restrict__ xh,
                             const float* __restrict__ dw_w, const float* __restrict__ dw_b,
                             const float* __restrict__ ln_g, const float* __restrict__ ln_b,
                             float* __restrict__ out) {
  __shared__ float sv[4][C_INC];
  __shared__ float smu[4], svar[4];
  const int c = threadIdx.x;           // 0..63
  const int y = threadIdx.y;           // 0..3
  const int pix = blockIdx.x * 4 + y;
  const int wq = pix & 63, hq = (pix >> 6) & 63, nq = pix >> 12;
  float s = dw_b[c];
#pragma unroll
  for (int kh = 0; kh < 3; ++kh) {
#pragma unroll
    for (int kw = 0; kw < 3; ++kw) {
      const int hh = hq + kh - 1, ww = wq + kw - 1;
      if ((unsigned)hh < H_IMG && (unsigned)ww < W_IMG)
        s += xh[(((size_t)nq << 12) + (hh << 6) + ww) * C_INC + c] *
             dw_w[(kh * 3 + kw) * C_INC + c];
    }
  }
  sv[y][c] = s;
  __syncthreads();
  if (c == 0) {
    float m = 0.f;
    for (int i = 0; i < C_INC; ++i) m += sv[y][i];
    m *= (1.0f / C_INC);
    float v = 0.f;
    for (int i = 0; i < C_INC; ++i) { const float d = sv[y][i] - m; v += d * d; }
    smu[y] = m; svar[y] = v * (1.0f / C_INC);
  }
  __syncthreads();
  const float nrm = ln_g[c] * (s - smu[y]) * rsqrtf(svar[y] + LN_EPS) + ln_b[c];
  out[(size_t)pix * C_INC + c] = 0.5f * nrm * (1.0f + erff(nrm * 0.70710678118654752f));
}

// ---------------------------------------------------------------------------
// 6) offset/mask projections + softmax; emit absolute padded-frame coords
// ---------------------------------------------------------------------------
__global__ void k_off_mask(const float* __restrict__ dw,
                           const float* __restrict__ off_w, const float* __restrict__ off_b,
                           const float* __restrict__ mask_w, const float* __restrict__ mask_b,
                           float* __restrict__ pxy, float* __restrict__ maskv) {
  __shared__ float sdw[C_INC];
  __shared__ float soff[72];
  __shared__ float smask[36];
  const int pix = blockIdx.x;
  const int tid = threadIdx.x;     // 128 threads
  if (tid < C_INC) sdw[tid] = dw[(size_t)pix * C_INC + tid];
  __syncthreads();
  if (tid < 72) {
    float a = off_b[tid];
    for (int k = 0; k < C_INC; ++k) a += sdw[k] * off_w[k * 72 + tid];
    soff[tid] = a;
  } else if (tid < 108) {
    const int m = tid - 72;
    float a = mask_b[m];
    for (int k = 0; k < C_INC; ++k) a += sdw[k] * mask_w[k * 36 + m];
    smask[m] = a;
  }
  __syncthreads();
  const int wq = pix & 63, hq = (pix >> 6) & 63;
  if (tid < 36) {                  // t = g*9 + p
    const int p = tid % 9;
    const float ox = soff[2 * tid + 0];
    const float oy = soff[2 * tid + 1];
    const float px = (float)wq + 1.0f + (float)(p / 3 - 1) + ox;  // padded frame
    const float py = (float)hq + 1.0f + (float)(p % 3 - 1) + oy;
    pxy[((size_t)pix * 36 + tid) * 2 + 0] = px;
    pxy[((size_t)pix * 36 + tid) * 2 + 1] = py;
  } else if (tid >= 64 && tid < 64 + G_GRP) {    // per-group softmax over 9
    const int g = tid - 64;
    float mx = -1e30f;
    for (int i = 0; i < 9; ++i) mx = fmaxf(mx, smask[g * 9 + i]);
    float sum = 0.f;
    for (int i = 0; i < 9; ++i) sum += expf(smask[g * 9 + i] - mx);
    const float inv = 1.0f / sum;
    for (int i = 0; i < 9; ++i)
      maskv[(size_t)pix * 36 + g * 9 + i] = expf(smask[g * 9 + i] - mx) * inv;
  }
}

// ---------------------------------------------------------------------------
// 7) deformable bilinear gather (zero-pad semantics) * mask, sum over P
// ---------------------------------------------------------------------------
__global__ void k_gather(const float* __restrict__ xp, const float* __restrict__ pxy,
                         const float* __restrict__ maskv, float* __restrict__ dcn) {
  const int c = threadIdx.x;       // 0..63
  const int g = c >> 4;
  const int pix = blockIdx.x;
  const int nq = pix >> 12;
  const float* xpn = xp + ((size_t)nq * HW_IMG) * C_INC;
  float acc = 0.f;
#pragma unroll
  for (int p = 0; p < 9; ++p) {
    const int t = g * 9 + p;
    const float fx = pxy[((size_t)pix * 36 + t) * 2 + 0] - 1.0f;  // -> unpadded frame
    const float fy = pxy[((size_t)pix * 36 + t) * 2 + 1] - 1.0f;
    const float x0f = floorf(fx), y0f = floorf(fy);
    const float wx = fx - x0f, wy = fy - y0f;
    const int x0 = (int)x0f, y0 = (int)y0f;
    float s = 0.f;
#pragma unroll
    for (int dy = 0; dy < 2; ++dy) {
#pragma unroll
      for (int dx = 0; dx < 2; ++dx) {
        const int yi = y0 + dy, xi = x0 + dx;
        if ((unsigned)yi < H_IMG && (unsigned)xi < W_IMG) {
          const float wgt = (dy ? wy : 1.0f - wy) * (dx ? wx : 1.0f - wx);
          s += wgt * xpn[((size_t)((yi << 6) + xi)) * C_INC + c];
        }
      }
    }
    acc += maskv[(size_t)pix * 36 + t] * s;
  }
  dcn[(size_t)pix * C_INC + c] = acc;
}

// ---------------------------------------------------------------------------
// 8) output_proj (WMMA) fused with BN1 + ReLU
// ---------------------------------------------------------------------------
__global__ void k_out_proj_bn1(const float* __restrict__ dcn, const float* __restrict__ w,
                               const float* __restrict__ bias,
                               const float* __restrict__ g1, const float* __restrict__ b1,
                               const float* __restrict__ m1, const float* __restrict__ v1,
                               float* __restrict__ y64) {
  const int wid  = blockIdx.x * (blockDim.x >> 5) + (threadIdx.x >> 5);
  const int tm   = wid >> 2;
  const int tn   = wid & 3;
  const int lane = threadIdx.x & 31;
  const int half = lane >> 4, l16 = lane & 15;
  const int ncol = tn * 16 + l16;
  const float bv = bias[ncol];
  v8f acc;
#pragma unroll
  for (int r = 0; r < 8; ++r) acc[r] = bv;
  acc = wmma_gemm_tile_k64(dcn + (size_t)tm * 16 * C_INC, C_INC, w + tn * 16, C_INC, acc);
  const float a1 = g1[ncol] * rsqrtf(v1[ncol] + BN_EPS);
  const float c1 = b1[ncol] - m1[ncol] * a1;
  float* out = y64 + (size_t)tm * 16 * C_INC + tn * 16 + l16;
#pragma unroll
  for (int r = 0; r < 8; ++r) out[(r + half * 8) * C_INC] = fmaxf(acc[r] * a1 + c1, 0.f);
}

// ---------------------------------------------------------------------------
// 9) conv1x1 64->128 (WMMA) fused with BN2 + ReLU, store NCHW to d_out
// ---------------------------------------------------------------------------
__global__ void k_conv_bn2(const float* __restrict__ y64, const float* __restrict__ w,
                           const float* __restrict__ bias,
                           const float* __restrict__ g2, const float* __restrict__ b2,
                           const float* __restrict__ m2, const float* __restrict__ v2,
                           float* __restrict__ out) {
  const int wid  = blockIdx.x * (blockDim.x >> 5) + (threadIdx.x >> 5);
  const int tm   = wid >> 3;      // 2048 tile rows
  const int tn   = wid & 7;       // 8 tile cols (128 channels)
  const int lane = threadIdx.x & 31;
  const int half = lane >> 4, l16 = lane & 15;
  const int ncol = tn * 16 + l16;
  const float bv = bias[ncol];
  v8f acc;
#pragma unroll
  for (int r = 0; r < 8; ++r) acc[r] = bv;
  acc = wmma_gemm_tile_k64(y64 + (size_t)tm * 16 * C_INC, C_INC, w + tn * 16, C_OUTC, acc);
  const float a2 = g2[ncol] * rsqrtf(v2[ncol] + BN_EPS);
  const float c2 = b2[ncol] - m2[ncol] * a2;
#pragma unroll
  for (int r = 0; r < 8; ++r) {
    const int pix = tm * 16 + r + half * 8;
    const int nq = pix >> 12;
    const int hw = pix & 4095;
    out[(((size_t)nq * C_OUTC + ncol) * HW_IMG) + hw] = fmaxf(acc[r] * a2 + c2, 0.f);
  }
}

// ---------------------------------------------------------------------------
extern "C" void kernel_launch(void* const* d_in, const int* in_sizes, int n_in,
                              void* d_out, int out_size, void* d_ws, size_t ws_size,
                              hipStream_t stream) {
  (void)in_sizes; (void)n_in; (void)out_size; (void)ws_size;
  const float* x      = (const float*)d_in[0];
  const float* refl   = (const float*)d_in[1];
  const float* gw1    = (const float*)d_in[2];
  const float* gb1    = (const float*)d_in[3];
  const float* gw2    = (const float*)d_in[4];
  const float* gb2    = (const float*)d_in[5];
  const float* tapw   = (const float*)d_in[6];
  const float* tapb   = (const float*)d_in[7];
  const float* dww    = (const float*)d_in[8];
  const float* dwbv   = (const float*)d_in[9];
  const float* lng    = (const float*)d_in[10];
  const float* lnb    = (const float*)d_in[11];
  const float* inpw   = (const float*)d_in[12];
  const float* inpb   = (const float*)d_in[13];
  const float* offw   = (const float*)d_in[14];
  const float* offb   = (const float*)d_in[15];
  const float* maskw  = (const float*)d_in[16];
  const float* maskb  = (const float*)d_in[17];
  const float* outpw  = (const float*)d_in[18];
  const float* outpb  = (const float*)d_in[19];
  const float* convw  = (const float*)d_in[20];
  const float* convb  = (const float*)d_in[21];
  const float* bn1g   = (const float*)d_in[22];
  const float* bn1b   = (const float*)d_in[23];
  const float* bn1m   = (const float*)d_in[24];
  const float* bn1v   = (const float*)d_in[25];
  const float* bn2g   = (const float*)d_in[26];
  const float* bn2b   = (const float*)d_in[27];
  const float* bn2m   = (const float*)d_in[28];
  const float* bn2v   = (const float*)d_in[29];

  float* ws     = (float*)d_ws;
  float* pooled = ws;                       // 512
  float* gain   = ws + 1024;                // 512
  float* xh     = ws + 4096;                // 2,097,152
  float* xp     = xh + (size_t)PIX_T * C_INC;
  float* dwact  = xp + (size_t)PIX_T * C_INC;
  float* pxy    = dwact + (size_t)PIX_T * C_INC;  // 2,359,296
  float* maskv  = pxy + (size_t)PIX_T * 72;       // 1,179,648
  float* dcn    = xh;      // xh dead after kernel 5
  float* y64    = dwact;   // dwact dead after kernel 6

  k_pool<<<N_IMG * C_INC, 256, 0, stream>>>(x, pooled);
  k_gain_mlp<<<1, 64, 0, stream>>>(pooled, gw1, gb1, gw2, gb2, gain);
  k_build_xh<<<(PIX_T * C_INC) / 256, 256, 0, stream>>>(x, refl, gain, tapw, tapb, xh);
  k_input_proj<<<1024, 256, 0, stream>>>(xh, inpw, inpb, xp);
  dim3 bE(64, 4);
  k_dw_ln_gelu<<<PIX_T / 4, bE, 0, stream>>>(xh, dww, dwbv, lng, lnb, dwact);
  k_off_mask<<<PIX_T, 128, 0, stream>>>(dwact, offw, offb, maskw, maskb, pxy, maskv);
  k_gather<<<PIX_T, 64, 0, stream>>>(xp, pxy, maskv, dcn);
  k_out_proj_bn1<<<1024, 256, 0, stream>>>(dcn, outpw, outpb, bn1g, bn1b, bn1m, bn1v, y64);
  k_conv_bn2<<<2048, 256, 0, stream>>>(y64, convw, convb, bn2g, bn2b, bn2m, bn2v,
                                       (float*)d_out);
}